// TopK_23270132809929
// MI455X (gfx1250) — compile-verified
//
#include <hip/hip_runtime.h>
#include <stdint.h>

#define NTHREADS 256
#define NCOLS    16384
#define EPT      (NCOLS / NTHREADS)      // 64 elements per thread
#define V4PT     (NCOLS / 4 / NTHREADS)  // 16 float4 per thread

typedef int v4i __attribute__((ext_vector_type(4)));
typedef __attribute__((address_space(1))) v4i gbl_v4i;
typedef __attribute__((address_space(3))) v4i lds_v4i;

__global__ __launch_bounds__(NTHREADS)
void topk_mask_kernel(const float* __restrict__ x,
                      const int*   __restrict__ kptr,
                      float*       __restrict__ out,
                      int ncols_chk)
{
    __shared__ float        rowbuf[NCOLS];
    __shared__ unsigned int hist[256];
    __shared__ unsigned int s_prefix, s_want, s_eq_taken;

    const int       tid  = threadIdx.x;
    const long long row  = blockIdx.x;
    const float* __restrict__ xrow = x   + row * (long long)NCOLS;
    float*       __restrict__ orow = out + row * (long long)NCOLS;
    (void)ncols_chk;

    // ---- Stage the entire row into LDS (async DMA path on gfx1250) ----
#if __has_builtin(__builtin_amdgcn_global_load_async_to_lds_b128)
    {
        float* xrow_nc = const_cast<float*>(xrow);
        for (int i = 0; i < V4PT; ++i) {
            const int v = i * NTHREADS + tid;             // float4 index, coalesced
            __builtin_amdgcn_global_load_async_to_lds_b128(
                (gbl_v4i*)(xrow_nc + (size_t)v * 4),
                (lds_v4i*)(&rowbuf[v * 4]),
                /*offset=*/0, /*cpol=*/0);
        }
    }
  #if __has_builtin(__builtin_amdgcn_s_wait_asynccnt)
    __builtin_amdgcn_s_wait_asynccnt(0);
  #else
    asm volatile("s_wait_asynccnt 0" ::: "memory");
  #endif
#else
    for (int i = 0; i < V4PT; ++i) {
        const int v = i * NTHREADS + tid;
        ((float4*)rowbuf)[v] = ((const float4*)xrow)[v];
    }
#endif
    __syncthreads();

    const unsigned int k = (unsigned int)(*kptr);

    // ---- Exact radix select (MSB-first, 8-bit digits) on key = abs bits ----
    // key = float bits & 0x7fffffff orders identically to |x| for finite inputs.
    unsigned int prefix = 0u;
    unsigned int want   = k;   // how many elements still to account for at/below current prefix

    for (int pass = 0; pass < 4; ++pass) {
        const int shift = 24 - 8 * pass;

        hist[tid & 255] = 0u;  // NTHREADS == 256: each thread clears one bin
        __syncthreads();

        const unsigned int himask = (pass == 0) ? 0u : (0xFFFFFFFFu << (shift + 8));
        const unsigned int hiref  = prefix & himask;

        for (int i = 0; i < EPT; ++i) {
            const int j = i * NTHREADS + tid;             // lane-consecutive: no LDS bank conflicts
            const unsigned int key =
                __float_as_uint(rowbuf[j]) & 0x7FFFFFFFu;
            if ((key & himask) == hiref) {
                atomicAdd(&hist[(key >> shift) & 0xFFu], 1u);
            }
        }
        __syncthreads();

        if (tid == 0) {
            unsigned int cum = 0u;
            int b = 255;
            for (; b > 0; --b) {
                const unsigned int h = hist[b];
                if (cum + h >= want) break;
                cum += h;
            }
            s_prefix = prefix | ((unsigned int)b << shift);
            s_want   = want - cum;     // count still needed within this digit bucket
        }
        __syncthreads();
        prefix = s_prefix;
        want   = s_want;
        // next iteration's hist clear is safe: everyone passed the barrier above
    }

    const unsigned int T   = prefix;   // exact k-th largest key
    const unsigned int rem = want;     // number of (key == T) elements to keep (==1 for distinct data)

    if (tid == 0) s_eq_taken = 0u;
    __syncthreads();

    // ---- Emit masked row: keep key > T always; keep first `rem` of key == T ----
    for (int i = 0; i < EPT; ++i) {
        const int j = i * NTHREADS + tid;                 // coalesced 128B/wave stores
        const float v = rowbuf[j];
        const unsigned int key = __float_as_uint(v) & 0x7FFFFFFFu;
        float o = 0.0f;
        if (key > T) {
            o = v;
        } else if (key == T) {
            const unsigned int slot = atomicAdd(&s_eq_taken, 1u);
            if (slot < rem) o = v;
        }
        orow[j] = o;
    }
}

extern "C" void kernel_launch(void* const* d_in, const int* in_sizes, int n_in,
                              void* d_out, int out_size, void* d_ws, size_t ws_size,
                              hipStream_t stream)
{
    (void)n_in; (void)d_ws; (void)ws_size; (void)out_size;
    const float* x    = (const float*)d_in[0];
    const int*   kptr = (const int*)d_in[1];
    float*       out  = (float*)d_out;

    const int rows = in_sizes[0] / NCOLS;   // 4096 for the reference shape

    topk_mask_kernel<<<dim3(rows), dim3(NTHREADS), 0, stream>>>(x, kptr, out, NCOLS);
}